// DETRLoss_76158360092707
// MI455X (gfx1250) — compile-verified
//
#include <hip/hip_runtime.h>
#include <hip/hip_bf16.h>
#include <stdint.h>

// DETR-style Hungarian matching loss for MI455X (gfx1250).
// One wave32 per batch: lane L owns Hungarian columns L+1 and L+33 (1-based),
// duals in f64 registers to bit-match the reference's float64 JV solve.
// Per-batch tiles are staged global->LDS with CDNA5 async-tensor-path
// instructions (global_load_async_to_lds_b128 / s_wait_asynccnt).

#define NS 64            // strokes / preds per batch
#define LANES 32
#define INF_D 1.0e30

__device__ __forceinline__ float wave_sum_f32(float v) {
#pragma unroll
  for (int off = 16; off > 0; off >>= 1) v += __shfl_xor(v, off, 32);
  return v;
}

__global__ __launch_bounds__(32)
void detr_hungarian_loss_kernel(const float* __restrict__ ps_g,   // [B,64,10]
                                const float* __restrict__ pv_g,   // [B,64,1]
                                const float* __restrict__ tg_g,   // [B,64,11]
                                float* __restrict__ loss_ws) {    // [B]
  __shared__ __align__(16) float sPS[NS * 10];   // pred strokes tile
  __shared__ __align__(16) float sTG[NS * 11];   // target tile
  __shared__ float  sC[NS * NS];                 // cost: [gt_compact][pred]
  __shared__ double sU[NS + 1];                  // row duals u[0..64]
  __shared__ int    sValid[NS];                  // compact gt -> original slot
  __shared__ int    sMatch[NS];                  // compact gt -> matched pred

  const int b    = blockIdx.x;
  const int lane = threadIdx.x;

  // ---- Stage per-batch tiles into LDS via CDNA5 async global->LDS DMA ----
  {
    unsigned long long gps = (unsigned long long)(ps_g + (size_t)b * (NS * 10));
    unsigned long long gtg = (unsigned long long)(tg_g + (size_t)b * (NS * 11));
    // Low 32 bits of a generic LDS pointer == wave-relative LDS byte offset.
    unsigned lps = (unsigned)(unsigned long long)(void*)sPS;
    unsigned ltg = (unsigned)(unsigned long long)(void*)sTG;
#pragma unroll
    for (int it = 0; it < 5; ++it) {             // 160 float4 = 2560 B
      unsigned voff = (unsigned)(lane + it * 32) * 16u;
      asm volatile("global_load_async_to_lds_b128 %0, %1, %2"
                   :: "v"(lps + voff), "v"(voff), "s"(gps) : "memory");
    }
#pragma unroll
    for (int it = 0; it < 6; ++it) {             // 176 float4 = 2816 B
      int idx = lane + it * 32;
      if (idx < (NS * 11) / 4) {
        unsigned voff = (unsigned)idx * 16u;
        asm volatile("global_load_async_to_lds_b128 %0, %1, %2"
                     :: "v"(ltg + voff), "v"(voff), "s"(gtg) : "memory");
      }
    }
    asm volatile("s_wait_asynccnt 0" ::: "memory");
  }
  __syncthreads();

  // ---- Compact valid targets (np.nonzero order) ----
  bool va = sTG[lane * 11 + 10] > 0.5f;
  bool vb = sTG[(lane + 32) * 11 + 10] > 0.5f;
  unsigned m0 = (unsigned)__ballot(va);
  unsigned m1 = (unsigned)__ballot(vb);
  const int ng = __popc(m0) + __popc(m1);
  unsigned lt = (1u << lane) - 1u;
  if (va) sValid[__popc(m0 & lt)] = lane;
  if (vb) sValid[__popc(m0) + __popc(m1 & lt)] = lane + 32;
  __syncthreads();

  // ---- Cost matrix: C[i][j] = 5*sum|d[:8]| + sum|d[8:10]|  (f32, as ref) ----
  for (int idx = lane; idx < ng * NS; idx += LANES) {
    int i = idx >> 6, j = idx & 63;
    const float* g = &sTG[sValid[i] * 11];
    const float* q = &sPS[j * 10];
    float cs = 0.f;
#pragma unroll
    for (int k = 0; k < 8; ++k) cs += fabsf(q[k] - g[k]);
    float cw = fabsf(q[8] - g[8]) + fabsf(q[9] - g[9]);
    sC[i * NS + j] = 5.0f * cs + cw;
  }
  sU[lane] = 0.0;
  sU[lane + 32] = 0.0;
  if (lane == 0) sU[64] = 0.0;
  __syncthreads();

  // ---- Jonker-Volgenant Hungarian: rows = compact gts (n=ng), cols = 64 preds
  // Lane-owned columns (1-based): jA = lane+1, jB = lane+33.
  const int jA = lane + 1, jB = lane + 33;
  double vA = 0.0, vB = 0.0;       // column duals
  int pA = 0, pB = 0;              // row matched to column (0 = none)

  for (int i = 1; i <= ng; ++i) {
    double mvA = INF_D, mvB = INF_D;
    int wayA = 0, wayB = 0;
    bool usedA = false, usedB = false;
    int j0 = 0;
    int i0 = i;                    // p[0] = i

    while (true) {
      __syncthreads();             // order prior sU writes before broadcast read
      double u0 = sU[i0];
      if (!usedA) {
        double cur = (double)sC[(i0 - 1) * NS + (jA - 1)] - u0 - vA;
        if (cur < mvA) { mvA = cur; wayA = j0; }
      }
      if (!usedB) {
        double cur = (double)sC[(i0 - 1) * NS + (jB - 1)] - u0 - vB;
        if (cur < mvB) { mvB = cur; wayB = j0; }
      }
      // argmin over unused columns; ties -> smallest column index (np.argmin)
      double cA = usedA ? (2.0 * INF_D) : mvA;
      double cB = usedB ? (2.0 * INF_D) : mvB;
      double bv; int bj;
      if (cB < cA) { bv = cB; bj = jB; } else { bv = cA; bj = jA; }
#pragma unroll
      for (int off = 16; off > 0; off >>= 1) {
        double ov = __shfl_xor(bv, off, 32);
        int    oj = __shfl_xor(bj, off, 32);
        if (ov < bv || (ov == bv && oj < bj)) { bv = ov; bj = oj; }
      }
      const double delta = bv;
      const int j1 = bj;

      // dual updates: u[p[used]] += delta (incl. u[i] for col 0), v[used] -= delta
      if (lane == 0) sU[i] += delta;
      if (usedA) { sU[pA] += delta; vA -= delta; } else { mvA -= delta; }
      if (usedB) { sU[pB] += delta; vB -= delta; } else { mvB -= delta; }

      j0 = j1;
      int owner = (j0 - 1) & 31;
      int slot  = (j0 - 1) >> 5;
      int pj0 = __shfl(slot ? pB : pA, owner, 32);
      if (pj0 == 0) break;         // reached a free column -> augment
      i0 = pj0;
      if (lane == owner) { if (slot) usedB = true; else usedA = true; }
    }

    // Augment: while j0: j1=way[j0]; p[j0]=p[j1]; j0=j1  (p[0]==i)
    while (true) {
      int owner = (j0 - 1) & 31;
      int slot  = (j0 - 1) >> 5;
      int jprev = __shfl(slot ? wayB : wayA, owner, 32);
      int newp;
      if (jprev == 0) {
        newp = i;
      } else {
        int o2 = (jprev - 1) & 31, s2 = (jprev - 1) >> 5;
        newp = __shfl(s2 ? pB : pA, o2, 32);
      }
      if (lane == owner) { if (slot) pB = newp; else pA = newp; }
      if (jprev == 0) break;
      j0 = jprev;
    }
  }

  // Invert matching: compact gt row -> pred column
  if (pA > 0) sMatch[pA - 1] = jA - 1;
  if (pB > 0) sMatch[pB - 1] = jB - 1;
  __syncthreads();

  // ---- Loss terms ----
  float coordS = 0.f, widthS = 0.f;
  for (int t = lane; t < ng; t += LANES) {
    int j = sMatch[t];
    const float* g = &sTG[sValid[t] * 11];
    const float* q = &sPS[j * 10];
#pragma unroll
    for (int k = 0; k < 8; ++k) coordS += fabsf(q[k] - g[k]);
    widthS += fabsf(q[8] - g[8]) + fabsf(q[9] - g[9]);
  }
  float bceS = 0.f;
  const float* pvb = pv_g + (size_t)b * NS;
#pragma unroll
  for (int it = 0; it < 2; ++it) {
    int s2 = lane + it * 32;
    float pv = pvb[s2];
    float gv = sTG[s2 * 11 + 10];
    float lp  = fmaxf(logf(pv), -100.f);
    float l1p = fmaxf(logf(1.f - pv), -100.f);
    bceS -= gv * lp + (1.f - gv) * l1p;
  }
  coordS = wave_sum_f32(coordS);
  widthS = wave_sum_f32(widthS);
  bceS   = wave_sum_f32(bceS);

  if (lane == 0) {
    float bce = bceS * (1.0f / (float)NS);
    float lb;
    if (ng > 0) {
      float ngf = (float)ng;
      lb = 5.0f * (coordS / (ngf * 8.0f)) + (widthS / (ngf * 2.0f)) + bce;
    } else {
      lb = bce;
    }
    loss_ws[b] = lb;
  }
}

// Deterministic fixed-order mean over batches (no float atomics).
__global__ __launch_bounds__(32)
void detr_loss_reduce_kernel(const float* __restrict__ ws, float* __restrict__ out,
                             int B) {
  int lane = threadIdx.x;
  float s = 0.f;
  for (int i = lane; i < B; i += 32) s += ws[i];
#pragma unroll
  for (int off = 16; off > 0; off >>= 1) s += __shfl_xor(s, off, 32);
  if (lane == 0) out[0] = s / (float)B;
}

extern "C" void kernel_launch(void* const* d_in, const int* in_sizes, int n_in,
                              void* d_out, int out_size, void* d_ws, size_t ws_size,
                              hipStream_t stream) {
  (void)n_in; (void)out_size; (void)ws_size;
  const float* ps = (const float*)d_in[0];   // pred_strokes [B,64,10]
  const float* pv = (const float*)d_in[1];   // pred_validity [B,64,1]
  const float* tg = (const float*)d_in[2];   // targets [B,64,11]
  float* ws  = (float*)d_ws;                 // [B] per-batch losses
  float* out = (float*)d_out;                // scalar

  const int B = in_sizes[0] / (NS * 10);     // 256

  detr_hungarian_loss_kernel<<<dim3(B), dim3(32), 0, stream>>>(ps, pv, tg, ws);
  detr_loss_reduce_kernel<<<dim3(1), dim3(32), 0, stream>>>(ws, out, B);
}